// NetGCN_72945724555676
// MI455X (gfx1250) — compile-verified
//
#include <hip/hip_runtime.h>
#include <hip/hip_bf16.h>

// ---------------------------------------------------------------------------
// NetGCN on MI455X (gfx1250): WMMA-f32 GEMMs + fused edge-MLP/message/scatter.
//
// Dims (fixed by reference): N=100000 nodes, E=1000000 edges, NODE_DIM=128,
// EDGE_DIM=16, HID=64, OUT=10, G=256 graphs.
//
// WMMA layout assumptions (CDNA5 ISA 7.12.2, 32-bit):
//   A (16x4, MxK):  lane<16:  a.x=A[m][k0],   a.y=A[m][k0+1]
//                   lane>=16: a.x=A[m][k0+2], a.y=A[m][k0+3]   (m = lane&15)
//   B (4x16, KxN):  mirrored: lane<16 holds K=k0,k0+1 at N=lane;
//                   lane>=16 holds K=k0+2,k0+3 at N=lane-16
//   C/D (16x16):    VGPR r: lane<16 -> (M=r, N=lane); lane>=16 -> (M=r+8, N=lane-16)
//
// Weights are staged into LDS once per block (We2 via GLOBAL_LOAD_ASYNC_TO_LDS,
// ASYNCcnt-tracked); waves then grid-stride over 16-row tiles so the staging
// cost is amortized and B-fragment reads come from LDS instead of VMEM.
// ---------------------------------------------------------------------------

typedef __attribute__((ext_vector_type(2))) float v2f;
typedef __attribute__((ext_vector_type(8))) float v8f;

#define NN 100000
#define EE 1000000
#define D0 128
#define HH 64
#define GG 256
#define ETILES (EE / 16)  // 62500

static __device__ __forceinline__ v8f wmma_f32_4(v2f a, v2f b, v8f c) {
  return __builtin_amdgcn_wmma_f32_16x16x4_f32(false, a, false, b, (short)0, c,
                                               false, false);
}

// Async global->LDS copy of 16 bytes per lane (gfx1250 GLOBAL_LOAD_ASYNC_TO_LDS_B128).
static __device__ __forceinline__ void async_copy_b128(unsigned lds_byte_addr,
                                                       const void* gaddr) {
  asm volatile("global_load_async_to_lds_b128 %0, %1, off" ::"v"(lds_byte_addr),
               "v"(gaddr));
}
static __device__ __forceinline__ void async_wait0() {
  asm volatile("s_wait_asynccnt 0x0" ::: "memory");
}

// ---------------------------------------------------------------------------
// Generic  out[M,64] = (relu?)(A[M,K] @ W[K,64] + bias[64]).  W staged in LDS,
// one wave per 16-row tile, waves grid-stride over tiles.  K<=256, K%4==0,
// M%16==0.  blockDim = 128 (4 waves).
// ---------------------------------------------------------------------------
__global__ void wmma_gemm_bias(const float* __restrict__ A,
                               const float* __restrict__ W,
                               const float* __restrict__ bias,
                               float* __restrict__ out, int M, int K,
                               int do_relu) {
  __shared__ float sW[256 * 64];  // up to K=256
  __shared__ float sb[64];
  for (int i = threadIdx.x; i < K * 64; i += 128) sW[i] = W[i];
  if (threadIdx.x < 64) sb[threadIdx.x] = bias[threadIdx.x];
  __syncthreads();

  const int lane = threadIdx.x & 31;
  const int wv = threadIdx.x >> 5;
  const int m = lane & 15;
  const bool hi = lane >= 16;
  const int mrow = hi ? 8 : 0;
  const int ntiles = M >> 4;
  const int nwaves = gridDim.x * 4;

  for (int tile = blockIdx.x * 4 + wv; tile < ntiles; tile += nwaves) {
    const int row0 = tile * 16;
    v8f acc[4] = {};
    for (int k0 = 0; k0 < K; k0 += 4) {
      const int ka = k0 + (hi ? 2 : 0);
      v2f a;
      a.x = A[(row0 + m) * K + ka];
      a.y = A[(row0 + m) * K + ka + 1];
#pragma unroll
      for (int nt = 0; nt < 4; ++nt) {
        v2f b;
        b.x = sW[ka * 64 + nt * 16 + m];
        b.y = sW[(ka + 1) * 64 + nt * 16 + m];
        acc[nt] = wmma_f32_4(a, b, acc[nt]);
      }
    }
#pragma unroll
    for (int nt = 0; nt < 4; ++nt) {
      const int n = nt * 16 + m;
      const float bb = sb[n];
#pragma unroll
      for (int r = 0; r < 8; ++r) {
        float v = acc[nt][r] + bb;
        if (do_relu) v = v > 0.f ? v : 0.f;
        out[(row0 + mrow + r) * 64 + n] = v;
      }
    }
  }
}

// ---------------------------------------------------------------------------
// Fused edge kernel: per wave-iteration, 16 edges.
//   u  = relu(ea[16x16] @ We1 + be1)          (WMMA, K=16)
//   e2 = u @ We2 + be2                         (WMMA, K=64, A re-laid via LDS)
//   msg = dinv[row]*dinv[col] * relu(h[row] + e2)
//   atomicAdd(agg[col], msg)
// Tile guard is wave-uniform -> EXEC stays all-ones around every WMMA.
// ---------------------------------------------------------------------------
__global__ void edge_mlp_scatter(const float* __restrict__ ea,
                                 const int* __restrict__ row,
                                 const int* __restrict__ col,
                                 const float* __restrict__ We1,
                                 const float* __restrict__ be1,
                                 const float* __restrict__ We2,
                                 const float* __restrict__ be2,
                                 const float* __restrict__ h,
                                 const float* __restrict__ dinv,
                                 float* __restrict__ agg) {
  __shared__ float sWe1[16 * 64];
  __shared__ float sWe2[64 * 64];
  __shared__ float sbe1[64];
  __shared__ float sbe2[64];
  __shared__ float sm[4][16 * 66];  // per-wave 16x64 tile, row stride 66 (pad)

  const int t = threadIdx.x;  // 128 threads
  // Stage We1 / biases with plain loads.
  for (int i = t; i < 16 * 64; i += 128) sWe1[i] = We1[i];
  if (t < 64) {
    sbe1[t] = be1[t];
    sbe2[t] = be2[t];
  }
  // Stage We2 (16 KB) via async global->LDS: 128 threads x 8 x 16B.
  {
    const unsigned lds0 = (unsigned)(size_t)(&sWe2[0]) + (unsigned)(t * 16);
    const char* g0 = (const char*)We2 + t * 16;
#pragma unroll
    for (int i = 0; i < 8; ++i)
      async_copy_b128(lds0 + i * 2048, g0 + i * 2048);
    async_wait0();  // per-wave ASYNCcnt drain before barrier
  }
  __syncthreads();

  const int lane = t & 31;
  const int wv = t >> 5;
  const int m = lane & 15;
  const bool hi = lane >= 16;
  const int mrow = hi ? 8 : 0;
  const int nwaves = gridDim.x * 4;

  for (int tile = blockIdx.x * 4 + wv; tile < ETILES; tile += nwaves) {
    const int e0 = tile * 16;

    // GEMM1: [16 edges x 16 edge-feat] @ We1[16 x 64]
    v8f acc[4] = {};
    for (int k0 = 0; k0 < 16; k0 += 4) {
      const int ka = k0 + (hi ? 2 : 0);
      v2f a;
      a.x = ea[(e0 + m) * 16 + ka];
      a.y = ea[(e0 + m) * 16 + ka + 1];
#pragma unroll
      for (int nt = 0; nt < 4; ++nt) {
        v2f b;
        b.x = sWe1[ka * 64 + nt * 16 + m];
        b.y = sWe1[(ka + 1) * 64 + nt * 16 + m];
        acc[nt] = wmma_f32_4(a, b, acc[nt]);
      }
    }
    // bias + relu, C-layout -> row-major LDS (per-wave region; same-wave DS
    // ops are in-order so no barrier needed)
#pragma unroll
    for (int nt = 0; nt < 4; ++nt) {
      const int n = nt * 16 + m;
      const float bb = sbe1[n];
#pragma unroll
      for (int r = 0; r < 8; ++r) {
        float v = acc[nt][r] + bb;
        sm[wv][(mrow + r) * 66 + n] = v > 0.f ? v : 0.f;
      }
    }

    // GEMM2: u[16x64] @ We2[64x64], A from LDS in A-layout
    v8f acc2[4] = {};
    for (int k0 = 0; k0 < 64; k0 += 4) {
      const int ka = k0 + (hi ? 2 : 0);
      v2f a;
      a.x = sm[wv][m * 66 + ka];
      a.y = sm[wv][m * 66 + ka + 1];
#pragma unroll
      for (int nt = 0; nt < 4; ++nt) {
        v2f b;
        b.x = sWe2[ka * 64 + nt * 16 + m];
        b.y = sWe2[(ka + 1) * 64 + nt * 16 + m];
        acc2[nt] = wmma_f32_4(a, b, acc2[nt]);
      }
    }

    // Fused message + scatter-add.  (nt, r) holds element (M=mrow+r, N=nt*16+m).
#pragma unroll
    for (int r = 0; r < 8; ++r) {
      const int eM = e0 + mrow + r;
      const int src = row[eM];
      const int dst = col[eM];
      const float nrm = dinv[src] * dinv[dst];
#pragma unroll
      for (int nt = 0; nt < 4; ++nt) {
        const int n = nt * 16 + m;
        float v = acc2[nt][r] + sbe2[n] + h[src * 64 + n];
        v = v > 0.f ? v : 0.f;
        atomicAdd(&agg[dst * 64 + n], nrm * v);
      }
    }
  }
}

// ---------------------------------------------------------------------------
// Small helper kernels
// ---------------------------------------------------------------------------
__global__ void fill_f32(float* __restrict__ p, long n, float v) {
  long i = (long)blockIdx.x * blockDim.x + threadIdx.x;
  if (i < n) p[i] = v;
}

__global__ void deg_accum(const int* __restrict__ row, float* __restrict__ deg,
                          int E) {
  int i = blockIdx.x * blockDim.x + threadIdx.x;
  if (i < E) atomicAdd(&deg[row[i]], 1.0f);
}

__global__ void dinv_k(const float* __restrict__ deg, float* __restrict__ dinv,
                       int N) {
  int i = blockIdx.x * blockDim.x + threadIdx.x;
  if (i < N) dinv[i] = rsqrtf(deg[i]);
}

// y = relu(agg + relu(h + root)/deg); writes y in-place over agg; accumulates
// per-column sum/sumsq via LDS reduction then 128 atomics per block.
__global__ void combine_stats(const float* __restrict__ h,
                              float* __restrict__ agg,
                              const float* __restrict__ root,
                              const float* __restrict__ deg,
                              float* __restrict__ stats, int N) {
  const int c = threadIdx.x & 63;
  const int rg = threadIdx.x >> 6;  // 0..3
  float s = 0.f, s2 = 0.f;
  const float rc = root[c];
  for (int node = blockIdx.x * 4 + rg; node < N; node += gridDim.x * 4) {
    const int idx = node * 64 + c;
    float hv = h[idx] + rc;
    hv = hv > 0.f ? hv : 0.f;
    float y = agg[idx] + hv / deg[node];
    y = y > 0.f ? y : 0.f;
    agg[idx] = y;
    s += y;
    s2 += y * y;
  }
  __shared__ float ls[256], ls2[256];
  ls[threadIdx.x] = s;
  ls2[threadIdx.x] = s2;
  __syncthreads();
  if (rg == 0) {
    s = ls[c] + ls[64 + c] + ls[128 + c] + ls[192 + c];
    s2 = ls2[c] + ls2[64 + c] + ls2[128 + c] + ls2[192 + c];
    atomicAdd(&stats[c], s);
    atomicAdd(&stats[64 + c], s2);
  }
}

__global__ void bn_finalize(const float* __restrict__ stats,
                            const float* __restrict__ gamma,
                            const float* __restrict__ beta,
                            float* __restrict__ ss, float invN) {
  const int c = threadIdx.x;  // 64 threads
  const float mu = stats[c] * invN;
  const float var = stats[64 + c] * invN - mu * mu;
  const float sc = gamma[c] * rsqrtf(var + 1e-5f);
  ss[c] = sc;
  ss[64 + c] = beta[c] - mu * sc;
}

__global__ void bn_normalize(const float* __restrict__ y,
                             const float* __restrict__ ss,
                             float* __restrict__ hout,
                             float* __restrict__ hcat, int layer, int N) {
  const int i = blockIdx.x * blockDim.x + threadIdx.x;
  if (i >= N * 64) return;
  const int node = i >> 6;
  const int c = i & 63;
  const float v = y[i] * ss[c] + ss[64 + c];
  hout[i] = v;
  hcat[node * 256 + layer * 64 + c] = v;
}

__global__ void pool_accum(const float* __restrict__ hcat,
                           const int* __restrict__ batch,
                           float* __restrict__ psum, float* __restrict__ pcnt,
                           int N) {
  const int node = blockIdx.x;  // one block per node, 256 threads = 256 cols
  const int g = batch[node];
  atomicAdd(&psum[g * 256 + threadIdx.x], hcat[node * 256 + threadIdx.x]);
  if (threadIdx.x == 0) atomicAdd(&pcnt[g], 1.0f);
}

__global__ void pool_div(float* __restrict__ psum,
                         const float* __restrict__ pcnt, int G) {
  const int i = blockIdx.x * blockDim.x + threadIdx.x;
  if (i >= G * 256) return;
  float c = pcnt[i >> 8];
  c = c > 1.f ? c : 1.f;
  psum[i] /= c;
}

__global__ void fc4_kernel(const float* __restrict__ z,
                           const float* __restrict__ W,
                           const float* __restrict__ b, float* __restrict__ out,
                           int G) {
  const int i = blockIdx.x * blockDim.x + threadIdx.x;
  if (i >= G * 10) return;
  const int g = i / 10;
  const int o = i - g * 10;
  float s = b[o];
#pragma unroll
  for (int k = 0; k < 64; ++k) s += z[g * 64 + k] * W[k * 10 + o];
  out[i] = s;
}

// ---------------------------------------------------------------------------
// Host orchestration
// ---------------------------------------------------------------------------
extern "C" void kernel_launch(void* const* d_in, const int* in_sizes, int n_in,
                              void* d_out, int out_size, void* d_ws,
                              size_t ws_size, hipStream_t stream) {
  (void)in_sizes;
  (void)n_in;
  (void)out_size;
  (void)ws_size;

  const float* x = (const float*)d_in[0];
  const int* row = (const int*)d_in[1];          // edge_index[0]
  const int* colp = ((const int*)d_in[1]) + EE;  // edge_index[1]
  const float* ea = (const float*)d_in[2];
  const int* batch = (const int*)d_in[3];
  // conv params: base 4 + layer*9; order: W,b,We1,be1,We2,be2,root,gamma,beta
  const float* fc1W = (const float*)d_in[40];
  const float* fc1b = (const float*)d_in[41];
  const float* fc4W = (const float*)d_in[42];
  const float* fc4b = (const float*)d_in[43];

  // Workspace carve-out (floats)
  float* hA = (float*)d_ws;            // [N,64]  node-transform output
  float* hB = hA + (long)NN * 64;      // [N,64]  normalized layer output
  float* agg = hB + (long)NN * 64;     // [N,64]  scatter target / y in-place
  float* hcat = agg + (long)NN * 64;   // [N,256]
  float* deg = hcat + (long)NN * 256;  // [N]
  float* dinv = deg + NN;              // [N]
  float* stats = dinv + NN;            // [128]
  float* ss = stats + 128;             // [128]
  float* psum = ss + 128;              // [256,256]
  float* pcnt = psum + GG * 256;       // [256]
  float* z = pcnt + GG;                // [256,64]

  // ---- degree (row-based) and dinv, once ----
  fill_f32<<<(NN + 255) / 256, 256, 0, stream>>>(deg, NN, 1.0f);
  deg_accum<<<(EE + 255) / 256, 256, 0, stream>>>(row, deg, EE);
  dinv_k<<<(NN + 255) / 256, 256, 0, stream>>>(deg, dinv, NN);

  const float* hin = x;
  int K = D0;
  for (int L = 0; L < 4; ++L) {
    const float* W = (const float*)d_in[4 + L * 9 + 0];
    const float* b = (const float*)d_in[4 + L * 9 + 1];
    const float* We1 = (const float*)d_in[4 + L * 9 + 2];
    const float* be1 = (const float*)d_in[4 + L * 9 + 3];
    const float* We2 = (const float*)d_in[4 + L * 9 + 4];
    const float* be2 = (const float*)d_in[4 + L * 9 + 5];
    const float* root = (const float*)d_in[4 + L * 9 + 6];
    const float* gamma = (const float*)d_in[4 + L * 9 + 7];
    const float* beta = (const float*)d_in[4 + L * 9 + 8];

    // h = hin @ W + b   (6250 wave-tiles, grid-strided, 4 waves/block)
    {
      int tiles = NN / 16;
      int blocks = (tiles + 3) / 4;
      if (blocks > 512) blocks = 512;
      wmma_gemm_bias<<<blocks, 128, 0, stream>>>(hin, W, b, hA, NN, K, 0);
    }
    // agg = 0 then fused edge MLP + message scatter (62500 tiles, 2048 waves)
    fill_f32<<<((long)NN * 64 + 255) / 256, 256, 0, stream>>>(
        agg, (long)NN * 64, 0.0f);
    edge_mlp_scatter<<<512, 128, 0, stream>>>(ea, row, colp, We1, be1, We2, be2,
                                              hA, dinv, agg);
    // combine + batchnorm stats
    fill_f32<<<1, 128, 0, stream>>>(stats, 128, 0.0f);
    combine_stats<<<512, 256, 0, stream>>>(hA, agg, root, deg, stats, NN);
    bn_finalize<<<1, 64, 0, stream>>>(stats, gamma, beta, ss, 1.0f / NN);
    bn_normalize<<<((long)NN * 64 + 255) / 256, 256, 0, stream>>>(agg, ss, hB,
                                                                  hcat, L, NN);
    hin = hB;
    K = HH;
  }

  // ---- global mean pool over sorted batch ----
  fill_f32<<<(GG * 256 + 255) / 256, 256, 0, stream>>>(psum, GG * 256, 0.0f);
  fill_f32<<<1, 256, 0, stream>>>(pcnt, GG, 0.0f);
  pool_accum<<<NN, 256, 0, stream>>>(hcat, batch, psum, pcnt, NN);
  pool_div<<<(GG * 256 + 255) / 256, 256, 0, stream>>>(psum, pcnt, GG);

  // ---- fc1 (WMMA, M=256, K=256) + relu, then fc4 ----
  wmma_gemm_bias<<<4, 128, 0, stream>>>(psum, fc1W, fc1b, z, GG, 256, 1);
  fc4_kernel<<<(GG * 10 + 255) / 256, 256, 0, stream>>>(z, fc4W, fc4b,
                                                        (float*)d_out, GG);
}